// REWAEncoder_84293028151328
// MI455X (gfx1250) — compile-verified
//
#include <hip/hip_runtime.h>

typedef __attribute__((ext_vector_type(2))) float v2f;
typedef __attribute__((ext_vector_type(8))) float v8f;

#define D_DIM 1024
#define M_DIM 128
#define KC 128
#define LDK 132   // padded LDS row stride (floats): conflict-free ds_load_b64
#define ROWS_PER_WAVE 32
#define WAVES_PER_BLOCK 4
#define BLOCK_THREADS (WAVES_PER_BLOCK * 32)

// ---------------------------------------------------------------------------
// Kernel 1: W' = (1/sqrt(128)) * H128 * W   (fold the Hadamard into W)
// grid = D_DIM/M_DIM blocks of 256 threads; each block transforms a
// 128(m) x 128(d) tile of W along the m axis in LDS.
// ---------------------------------------------------------------------------
__global__ __launch_bounds__(256) void hadamard_w(const float* __restrict__ W,
                                                  float* __restrict__ Wh) {
  __shared__ float tile[M_DIM][M_DIM + 1];
  const int tid = threadIdx.x;
  const int col0 = blockIdx.x * M_DIM;

  for (int idx = tid; idx < M_DIM * M_DIM; idx += 256) {
    const int m = idx >> 7, c = idx & 127;
    tile[m][c] = W[(size_t)m * D_DIM + col0 + c];
  }
  __syncthreads();

  // Sylvester-ordered FWHT along m (matches reference butterfly exactly)
  for (int half = 1; half < M_DIM; half <<= 1) {
    for (int p = tid; p < (M_DIM / 2) * M_DIM; p += 256) {
      const int c = p & 127;
      const int pr = p >> 7;               // pair id 0..63 within a column
      const int g = pr / half;
      const int off = pr - g * half;
      const int i = g * 2 * half + off;
      const int j = i + half;
      const float a = tile[i][c];
      const float b = tile[j][c];
      tile[i][c] = a + b;
      tile[j][c] = a - b;
    }
    __syncthreads();
  }

  const float s = 0.08838834764831845f;    // 1/sqrt(128)
  for (int idx = tid; idx < M_DIM * M_DIM; idx += 256) {
    const int m = idx >> 7, c = idx & 127;
    Wh[(size_t)m * D_DIM + col0 + c] = tile[m][c] * s;
  }
}

// ---------------------------------------------------------------------------
// Kernel 2: out[row, m] = (sum_k x[row,k] * Wh[m,k]) > 0 ? 1 : 0
// One wave32 per 32-row x 128-col output slab using V_WMMA_F32_16X16X4_F32.
// Two 16-row tiles per wave -> every B fragment loaded from LDS feeds 2 wmma
// (halves LDS traffic vs 16-row slabs; LDS time ~ HBM floor ~6.5us).
//   A (16x4 f32): lane<16 holds {K,K+1}, lane>=16 holds {K+2,K+3} of row lane%16
//   B (4x16 f32): lane<16 holds {K,K+1}, lane>=16 holds {K+2,K+3} of col lane%16
//   C/D (16x16 f32): VGPR v, lane l -> M = v + 8*(l>>4), N = l&15
// ---------------------------------------------------------------------------
__global__ __launch_bounds__(BLOCK_THREADS) void gemm_fwht_sign(
    const float* __restrict__ x, const float* __restrict__ Wh,
    float* __restrict__ out) {
  __shared__ float bsm[M_DIM * LDK];       // 128*132*4 = 67,584 B
  const int tid = threadIdx.x;
  const int wave = tid >> 5;
  const int lane = tid & 31;
  const int mrow = lane & 15;
  const int hi = lane >> 4;
  const int khalf = hi * 2;
  const size_t rowBase =
      ((size_t)blockIdx.x * WAVES_PER_BLOCK + wave) * ROWS_PER_WAVE;
  const float* __restrict__ xrow0 = x + (rowBase + mrow) * (size_t)D_DIM;
  const float* __restrict__ xrow1 = x + (rowBase + 16 + mrow) * (size_t)D_DIM;

  v8f acc[2][8];
#pragma unroll
  for (int r = 0; r < 2; ++r)
#pragma unroll
    for (int n = 0; n < 8; ++n) {
      v8f z = {0.f, 0.f, 0.f, 0.f, 0.f, 0.f, 0.f, 0.f};
      acc[r][n] = z;
    }

  for (int k0 = 0; k0 < D_DIM; k0 += KC) {
    __syncthreads();                       // protect previous chunk's reads
    // Cooperative, coalesced load of Wh[:, k0:k0+KC] into padded LDS
    for (int idx = tid; idx < M_DIM * (KC / 4); idx += BLOCK_THREADS) {
      const int m = idx >> 5;
      const int c4 = (idx & 31) << 2;
      const float4 v = *(const float4*)(Wh + (size_t)m * D_DIM + k0 + c4);
      float* dst = &bsm[m * LDK + c4];
      dst[0] = v.x; dst[1] = v.y; dst[2] = v.z; dst[3] = v.w;
    }
    __syncthreads();

    // Prefetch next x K-chunk into cache while we grind on this one
    if (k0 + KC < D_DIM) {
      __builtin_prefetch(xrow0 + k0 + KC + khalf, 0, 0);
      __builtin_prefetch(xrow1 + k0 + KC + khalf, 0, 0);
    }

#pragma unroll 2
    for (int kk = 0; kk < KC; kk += 4) {
      // A fragments: 2 consecutive floats of each row tile's x row
      const float2 af0 = *(const float2*)(xrow0 + k0 + kk + khalf);
      const float2 af1 = *(const float2*)(xrow1 + k0 + kk + khalf);
      v2f a0; a0.x = af0.x; a0.y = af0.y;
      v2f a1; a1.x = af1.x; a1.y = af1.y;
#pragma unroll
      for (int n = 0; n < 8; ++n) {
        const float* bp = &bsm[(n * 16 + mrow) * LDK + kk + khalf];
        v2f b; b.x = bp[0]; b.y = bp[1];
        // (neg_a, A, neg_b, B, c_mod, C, reuse_a, reuse_b)
        acc[0][n] = __builtin_amdgcn_wmma_f32_16x16x4_f32(
            false, a0, false, b, (short)0, acc[0][n], false, false);
        acc[1][n] = __builtin_amdgcn_wmma_f32_16x16x4_f32(
            false, a1, false, b, (short)0, acc[1][n], false, false);
      }
    }
  }

  // Emit bits: out[rowBase + r*16 + v + 8*hi, n*16 + mrow] = acc[r][n][v] > 0
#pragma unroll
  for (int r = 0; r < 2; ++r) {
    float* __restrict__ obase = out + (rowBase + r * 16) * (size_t)M_DIM;
#pragma unroll
    for (int n = 0; n < 8; ++n) {
#pragma unroll
      for (int v = 0; v < 8; ++v) {
        const size_t off = (size_t)(v + 8 * hi) * M_DIM + n * 16 + mrow;
        obase[off] = acc[r][n][v] > 0.0f ? 1.0f : 0.0f;
      }
    }
  }
}

// ---------------------------------------------------------------------------
extern "C" void kernel_launch(void* const* d_in, const int* in_sizes, int n_in,
                              void* d_out, int out_size, void* d_ws, size_t ws_size,
                              hipStream_t stream) {
  const float* x = (const float*)d_in[0];   // [4, 8192, 1024] f32
  const float* W = (const float*)d_in[1];   // [128, 1024] f32
  float* out = (float*)d_out;               // [4, 8192, 128] f32
  float* Wh = (float*)d_ws;                 // 128*1024 f32 = 512 KB scratch

  hadamard_w<<<D_DIM / M_DIM, 256, 0, stream>>>(W, Wh);

  const int total_rows = in_sizes[0] / D_DIM;                  // B*N = 32768
  const int rows_per_block = WAVES_PER_BLOCK * ROWS_PER_WAVE;  // 128
  gemm_fwht_sign<<<total_rows / rows_per_block, BLOCK_THREADS, 0, stream>>>(
      x, Wh, out);
}